// HungarianMatcher_62371515072731
// MI455X (gfx1250) — compile-verified
//
#include <hip/hip_runtime.h>
#include <hip/hip_bf16.h>

typedef __attribute__((ext_vector_type(2))) float v2f;
typedef __attribute__((ext_vector_type(8))) float v8f;

#define ALPHA   0.25f
#define EPS     1e-8f
#define W_CLASS 2.0f
#define W_BBOX  5.0f
#define W_GIOU  2.0f
#define KPAD    96   // 91 classes padded to multiple of 4 (and nice 384B row stride)

// ---------------------------------------------------------------------------
// Kernel 1: per-query focal-loss class-cost table T[n, c], c in [0, KPAD)
// T[n,c] = alpha*(1-p)^2*(-log(p+eps)) - (1-alpha)*p^2*(-log(1-p+eps))
// Padding columns are 0 so the WMMA dot over K=96 is exact.
// ---------------------------------------------------------------------------
__global__ void class_table_kernel(const float* __restrict__ logits,
                                   float* __restrict__ T, int N, int C) {
    int idx = blockIdx.x * blockDim.x + threadIdx.x;
    if (idx >= N * KPAD) return;
    int n = idx / KPAD;
    int c = idx - n * KPAD;
    float v = 0.0f;
    if (c < C) {
        float x = logits[n * C + c];
        float p = 1.0f / (1.0f + __expf(-x));
        float q = 1.0f - p;
        float pos = ALPHA * q * q * (-__logf(p + EPS));
        float neg = (1.0f - ALPHA) * p * p * (-__logf(q + EPS));
        v = pos - neg;
    }
    T[idx] = v;
}

// ---------------------------------------------------------------------------
// Kernel 2: transposed scaled one-hot  Ht[m, c] = (label[m]==c) ? W_CLASS : 0
// (rows m >= M are all-zero so padded output columns accumulate 0)
// ---------------------------------------------------------------------------
__global__ void onehot_kernel(const int* __restrict__ labels,
                              float* __restrict__ Ht, int M, int Mpad) {
    int idx = blockIdx.x * blockDim.x + threadIdx.x;
    if (idx >= Mpad * KPAD) return;
    int m = idx / KPAD;
    int c = idx - m * KPAD;
    float v = 0.0f;
    if (m < M && labels[m] == c) v = W_CLASS;
    Ht[idx] = v;
}

// ---------------------------------------------------------------------------
// Kernel 3: padded raw GT boxes (cx,cy,w,h); pads get w=h=1 to avoid 0/0.
// ---------------------------------------------------------------------------
__global__ void gtpad_kernel(const float* __restrict__ gt,
                             float* __restrict__ Gp, int M, int Mpad) {
    int m = blockIdx.x * blockDim.x + threadIdx.x;
    if (m >= Mpad) return;
    float4 b = (m < M) ? ((const float4*)gt)[m] : make_float4(0.f, 0.f, 1.f, 1.f);
    ((float4*)Gp)[m] = b;
}

// ---------------------------------------------------------------------------
// Kernel 4: main cost kernel. One wave owns a 16x16 output tile.
//   - class term: 24 chained v_wmma_f32_16x16x4_f32 over K=96 (exact gather,
//     W_CLASS pre-folded into the one-hot B operand)
//   - bbox L1 + GIoU terms: VALU over the accumulator layout
//     (VGPR r, lane l) -> row r + 8*(l>=16), col l&15
// ---------------------------------------------------------------------------
__global__ void __launch_bounds__(256)
cost_kernel(const float* __restrict__ T,
            const float* __restrict__ Ht,
            const float* __restrict__ Gp,
            const float* __restrict__ pboxes,
            float* __restrict__ out,
            int M, int CT, int NT) {
    int wave = threadIdx.x >> 5;
    int lane = threadIdx.x & 31;
    int tile = blockIdx.x * (blockDim.x >> 5) + wave;
    if (tile >= NT) return;               // wave-uniform: EXEC stays all-1s

    int rt = tile / CT;
    int ct = tile - rt * CT;
    int qbase = rt << 4;
    int cbase = ct << 4;
    int lo = lane & 15;
    int hi = lane >> 4;

    // A layout (16x4 f32): lane<16 holds (M=lo,K=k0+{0,1}), lane>=16 (M=lo,K=k0+{2,3})
    // B layout (4x16 f32): lane<16 holds (K=k0+{0,1},N=lo) via transposed Ht rows
    const float* Arow = T  + (size_t)(qbase + lo) * KPAD + 2 * hi;
    const float* Brow = Ht + (size_t)(cbase + lo) * KPAD + 2 * hi;

    v8f acc = {};
#pragma unroll
    for (int k0 = 0; k0 < KPAD; k0 += 4) {
        v2f a = *(const v2f*)(Arow + k0);
        v2f b = *(const v2f*)(Brow + k0);
        acc = __builtin_amdgcn_wmma_f32_16x16x4_f32(
            /*neg_a=*/false, a, /*neg_b=*/false, b,
            /*c_mod=*/(short)0, acc, /*reuse_a=*/false, /*reuse_b=*/false);
    }

    // GT box for this lane's column: loop-invariant over the 8 row elements.
    int g = cbase + lo;
    float4 gb = ((const float4*)Gp)[g];
    float gx0 = gb.x - 0.5f * gb.z, gy0 = gb.y - 0.5f * gb.w;
    float gx1 = gb.x + 0.5f * gb.z, gy1 = gb.y + 0.5f * gb.w;
    float garea = (gx1 - gx0) * (gy1 - gy0);

#pragma unroll
    for (int r = 0; r < 8; ++r) {
        int q = qbase + r + 8 * hi;
        float4 pb = ((const float4*)pboxes)[q];

        // L1 over raw (cx,cy,w,h)
        float c_bbox = fabsf(pb.x - gb.x) + fabsf(pb.y - gb.y) +
                       fabsf(pb.z - gb.z) + fabsf(pb.w - gb.w);

        // GIoU on xyxy
        float px0 = pb.x - 0.5f * pb.z, py0 = pb.y - 0.5f * pb.w;
        float px1 = pb.x + 0.5f * pb.z, py1 = pb.y + 0.5f * pb.w;
        float parea = (px1 - px0) * (py1 - py0);
        float iw = fmaxf(fminf(px1, gx1) - fmaxf(px0, gx0), 0.0f);
        float ih = fmaxf(fminf(py1, gy1) - fmaxf(py0, gy0), 0.0f);
        float inter = iw * ih;
        float uni = parea + garea - inter;
        float iou = inter / uni;
        float ew = fmaxf(fmaxf(px1, gx1) - fminf(px0, gx0), 0.0f);
        float eh = fmaxf(fmaxf(py1, gy1) - fminf(py0, gy0), 0.0f);
        float earea = ew * eh;
        float giou = iou - (earea - uni) / earea;

        float v = acc[r] + W_BBOX * c_bbox - W_GIOU * giou;  // acc = W_CLASS*class
        if (g < M) out[(size_t)q * M + g] = v;               // mask padded columns
    }
}

// ---------------------------------------------------------------------------
extern "C" void kernel_launch(void* const* d_in, const int* in_sizes, int n_in,
                              void* d_out, int out_size, void* d_ws, size_t ws_size,
                              hipStream_t stream) {
    const float* logits = (const float*)d_in[0];   // [16,900,91] f32
    const float* pboxes = (const float*)d_in[1];   // [16,900,4]  f32
    const float* gt     = (const float*)d_in[2];   // [1000,4]    f32
    const int*   labels = (const int*)d_in[3];     // [1000]      int
    float* out = (float*)d_out;                    // [16,900,1000] f32

    int Nq = in_sizes[1] / 4;            // 14400 (multiple of 16)
    int M  = in_sizes[2] / 4;            // 1000
    int C  = in_sizes[0] / Nq;           // 91
    int Mpad = ((M + 15) / 16) * 16;     // 1008
    int CT = Mpad / 16;                  // 63 column tiles
    int RT = Nq / 16;                    // 900 row tiles
    int NT = RT * CT;                    // 56700 wave-tiles

    // Workspace layout (~5.94 MB)
    float* T  = (float*)d_ws;                       // Nq   * KPAD
    float* Ht = T + (size_t)Nq * KPAD;              // Mpad * KPAD
    float* Gp = Ht + (size_t)Mpad * KPAD;           // Mpad * 4

    int t1 = Nq * KPAD;
    class_table_kernel<<<(t1 + 255) / 256, 256, 0, stream>>>(logits, T, Nq, C);
    int t2 = Mpad * KPAD;
    onehot_kernel<<<(t2 + 255) / 256, 256, 0, stream>>>(labels, Ht, M, Mpad);
    gtpad_kernel<<<(Mpad + 255) / 256, 256, 0, stream>>>(gt, Gp, M, Mpad);

    const int wavesPerBlock = 8;                    // 256 threads = 8 wave32
    int blocks = (NT + wavesPerBlock - 1) / wavesPerBlock;
    cost_kernel<<<blocks, 32 * wavesPerBlock, 0, stream>>>(T, Ht, Gp, pboxes,
                                                           out, M, CT, NT);
}